// GNNBlock_22771916604217
// MI455X (gfx1250) — compile-verified
//
#include <hip/hip_runtime.h>
#include <hip/hip_bf16.h>
#include <math.h>

typedef float v2f __attribute__((ext_vector_type(2)));
typedef float v8f __attribute__((ext_vector_type(8)));

#define D 128
#define DP (D + 4)     // LDS row padding: bank stride 132 % 64 = 4 -> conflict-free column access
#define NT 4           // row tiles per block (64 nodes/block) -> 4x weight-traffic reuse

// ---------------------------------------------------------------------------
// K0: zero workspace (agg, deg, colsum, colsumsq are contiguous)
// ---------------------------------------------------------------------------
__global__ void zero_ws_kernel(float* __restrict__ p, size_t n) {
    size_t i = (size_t)blockIdx.x * blockDim.x + threadIdx.x;
    if (i < n) p[i] = 0.0f;
}

// ---------------------------------------------------------------------------
// K1: edge scatter-add. One wave32 per edge: 128 channels = 4 floats/lane.
// agg[dst] += x[src]; deg[dst] += 1
// ---------------------------------------------------------------------------
__global__ __launch_bounds__(256) void edge_scatter_kernel(
    const float* __restrict__ x,
    const int*   __restrict__ src,
    const int*   __restrict__ dst,
    float* __restrict__ agg,
    float* __restrict__ deg,
    int E)
{
    int wave = blockIdx.x * (blockDim.x >> 5) + (threadIdx.x >> 5);
    int lane = threadIdx.x & 31;
    if (wave >= E) return;
    int s = src[wave];
    int d = dst[wave];
    const float4 v = ((const float4*)(x + (size_t)s * D))[lane];
    float* ad = agg + (size_t)d * D + lane * 4;
    atomicAdd(&ad[0], v.x);
    atomicAdd(&ad[1], v.y);
    atomicAdd(&ad[2], v.z);
    atomicAdd(&ad[3], v.w);
    if (lane == 0) atomicAdd(&deg[d], 1.0f);
}

// ---------------------------------------------------------------------------
// K2: fused SAGE linear (WMMA f32 16x16x4) + bias + row L2 normalize +
//     GraphNorm column statistics. Block = 256 threads = 8 wave32.
//     Each wave owns one 16-wide column tile; weight fragments live in
//     registers and are reused across NT=4 row tiles (64 nodes per block).
// ---------------------------------------------------------------------------
__global__ __launch_bounds__(256) void sage_gemm_kernel(
    const float* __restrict__ x,
    const float* __restrict__ agg,
    const float* __restrict__ deg,
    const float* __restrict__ Wl,   // [D,D] row-major: Wl[j*D+k]
    const float* __restrict__ bl,   // [D]
    const float* __restrict__ Wr,   // [D,D]
    float* __restrict__ h,          // [N,D] output (pre-GraphNorm)
    float* __restrict__ colsum,     // [D]
    float* __restrict__ colsumsq,   // [D]
    int n)
{
    __shared__ float As[16][DP];    // mean-aggregated neighbor features
    __shared__ float Xs[16][DP];    // root features
    __shared__ float Hs[16][DP];    // output tile
    __shared__ float rowred[16][16];
    __shared__ float rnorm[16];
    __shared__ float cpart[2][D];

    const int tid   = threadIdx.x;
    const int base0 = blockIdx.x * (16 * NT);

    // ---- per-wave geometry (ISA 7.12.2 fragment layouts) ----
    const int wv    = tid >> 5;          // wave -> column tile
    const int lane  = tid & 31;
    const int col0  = wv * 16;
    const int lcol  = lane & 15;
    const int arow  = lcol;              // A-frag: M = lane%16
    const int kbase = (lane & 16) ? 2 : 0;
    const int mrow  = (lane & 16) ? 8 : 0;

    // ---- hoist weight fragments into registers (reused for all row tiles) ----
    const float* WlC = Wl + (size_t)(col0 + lcol) * D;   // B[k][j] = Wl[j*D+k]
    const float* WrC = Wr + (size_t)(col0 + lcol) * D;
    v2f wlf[D / 4], wrf[D / 4];
#pragma unroll
    for (int kk = 0; kk < D / 4; ++kk) {
        int k0 = kk * 4 + kbase;
        wlf[kk] = (v2f){ WlC[k0], WlC[k0 + 1] };
        wrf[kk] = (v2f){ WrC[k0], WrC[k0 + 1] };
    }
    const float bj = bl[col0 + lcol];

    if (tid < D) { cpart[0][tid] = 0.0f; cpart[1][tid] = 0.0f; }

    for (int t = 0; t < NT; ++t) {
        const int row0 = base0 + t * 16;

        // -- cooperative tile load: 16 rows x 128 cols as float4 --
        for (int i = tid; i < 16 * (D / 4); i += 256) {
            int r  = i >> 5;          // 32 float4 per row
            int c4 = i & 31;
            int g  = row0 + r;
            float4 xa = {0.f, 0.f, 0.f, 0.f};
            float4 aa = {0.f, 0.f, 0.f, 0.f};
            float inv = 1.0f;
            if (g < n) {
                xa = ((const float4*)(x   + (size_t)g * D))[c4];
                aa = ((const float4*)(agg + (size_t)g * D))[c4];
                inv = 1.0f / fmaxf(deg[g], 1.0f);
            }
            ((float4*)&Xs[r][0])[c4] = xa;
            aa.x *= inv; aa.y *= inv; aa.z *= inv; aa.w *= inv;
            ((float4*)&As[r][0])[c4] = aa;
        }
        __syncthreads();

        // -- WMMA: h_tile = A @ Wl^T + X @ Wr^T --
        v8f acc = {};
#if __has_builtin(__builtin_amdgcn_wmma_f32_16x16x4_f32)
#pragma unroll
        for (int kk = 0; kk < D / 4; ++kk) {
            int k0 = kk * 4 + kbase;
            v2f a = { As[arow][k0], As[arow][k0 + 1] };
            acc = __builtin_amdgcn_wmma_f32_16x16x4_f32(false, a, false, wlf[kk],
                                                        (short)0, acc, false, false);
        }
#pragma unroll
        for (int kk = 0; kk < D / 4; ++kk) {
            int k0 = kk * 4 + kbase;
            v2f a = { Xs[arow][k0], Xs[arow][k0 + 1] };
            acc = __builtin_amdgcn_wmma_f32_16x16x4_f32(false, a, false, wrf[kk],
                                                        (short)0, acc, false, false);
        }
#else
        for (int i = 0; i < 8; ++i) {
            float s = 0.0f;
            int r = mrow + i;
            for (int k = 0; k < D; ++k)
                s += As[r][k] * WlC[k] + Xs[r][k] * WrC[k];
            acc[i] = s;
        }
#endif

        // C/D layout: VGPR i -> row M = i + (lane>=16 ? 8 : 0), col = col0 + lane%16
#pragma unroll
        for (int i = 0; i < 8; ++i)
            Hs[mrow + i][col0 + lcol] = acc[i] + bj;
        __syncthreads();

        // -- row L2 norm: 16 threads per row, 8 cols each --
        const int rr = tid >> 4;
        const int cc = (tid & 15) * 8;
        float ss = 0.0f;
#pragma unroll
        for (int i = 0; i < 8; ++i) { float v = Hs[rr][cc + i]; ss += v * v; }
        rowred[rr][tid & 15] = ss;
        __syncthreads();
        if (tid < 16) {
            float s = 0.0f;
#pragma unroll
            for (int i = 0; i < 16; ++i) s += rowred[tid][i];
            rnorm[tid] = 1.0f / fmaxf(sqrtf(s), 1e-12f);
        }
        __syncthreads();

        // -- scale, emit h, accumulate GraphNorm column sums (ds_add_f32);
        //    out-of-range rows must NOT pollute the statistics --
        const int g = row0 + rr;
        if (g < n) {
            const float inv = rnorm[rr];
            float vals[8];
#pragma unroll
            for (int i = 0; i < 8; ++i) {
                float v = Hs[rr][cc + i] * inv;
                vals[i] = v;
                atomicAdd(&cpart[0][cc + i], v);
                atomicAdd(&cpart[1][cc + i], v * v);
            }
            float4* hp = (float4*)(h + (size_t)g * D + cc);
            hp[0] = make_float4(vals[0], vals[1], vals[2], vals[3]);
            hp[1] = make_float4(vals[4], vals[5], vals[6], vals[7]);
        }
        __syncthreads();   // also protects As/Xs/Hs reuse in next tile
    }

    // -- single flush of column partials per block --
    if (tid < D) {
        atomicAdd(&colsum[tid],   cpart[0][tid]);
        atomicAdd(&colsumsq[tid], cpart[1][tid]);
    }
}

// ---------------------------------------------------------------------------
// K3: finalize GraphNorm stats.
// var = E[h^2] - 2*a*mu^2 + a^2*mu^2  (since centered = h - a*mu)
// ---------------------------------------------------------------------------
__global__ void gn_stats_kernel(const float* __restrict__ colsum,
                                const float* __restrict__ colsumsq,
                                const float* __restrict__ alpha,
                                float* __restrict__ mean,
                                float* __restrict__ invstd,
                                int n)
{
    int j = threadIdx.x;
    if (j < D) {
        float invn = 1.0f / (float)n;
        float m  = colsum[j] * invn;
        float e2 = colsumsq[j] * invn;
        float a  = alpha[j];
        float var = e2 - 2.0f * a * m * m + a * a * m * m;
        mean[j]   = m;
        invstd[j] = rsqrtf(var + 1e-5f);
    }
}

// ---------------------------------------------------------------------------
// K4: GraphNorm affine + exact GELU + residual (in-place on d_out)
// ---------------------------------------------------------------------------
__global__ __launch_bounds__(256) void gn_gelu_res_kernel(
    const float* __restrict__ x,
    float* __restrict__ out,             // holds h; rewritten with final result
    const float* __restrict__ gw,
    const float* __restrict__ gb,
    const float* __restrict__ alpha,
    const float* __restrict__ mean,
    const float* __restrict__ invstd,
    size_t total)
{
    size_t i = (size_t)blockIdx.x * blockDim.x + threadIdx.x;
    if (i >= total) return;
    int j = (int)(i & (D - 1));
    float hv = out[i];
    float c  = (hv - alpha[j] * mean[j]) * invstd[j];
    float gz = gw[j] * c + gb[j];
    float ge = 0.5f * gz * (1.0f + erff(gz * 0.70710678118654752f));
    out[i] = ge + x[i];
}

// ---------------------------------------------------------------------------
extern "C" void kernel_launch(void* const* d_in, const int* in_sizes, int n_in,
                              void* d_out, int out_size, void* d_ws, size_t ws_size,
                              hipStream_t stream)
{
    const float* x     = (const float*)d_in[0];
    const int*   ei    = (const int*)  d_in[1];   // [2,E]: row0=src, row1=dst
    // d_in[2] = batch (all zeros) -> unused
    const float* Wl    = (const float*)d_in[3];
    const float* bl    = (const float*)d_in[4];
    const float* Wr    = (const float*)d_in[5];
    const float* gw    = (const float*)d_in[6];
    const float* gb    = (const float*)d_in[7];
    const float* alpha = (const float*)d_in[8];

    const int n = in_sizes[0] / D;
    const int E = in_sizes[1] / 2;

    // workspace layout
    float* agg      = (float*)d_ws;                 // n*D
    float* deg      = agg + (size_t)n * D;          // n
    float* colsum   = deg + n;                      // D
    float* colsumsq = colsum + D;                   // D
    float* mean     = colsumsq + D;                 // D
    float* invstd   = mean + D;                     // D
    float* h        = (float*)d_out;                // n*D (pre-norm h, then final)

    const int* src = ei;
    const int* dst = ei + E;

    // K0: zero agg+deg+colsum+colsumsq (contiguous)
    size_t zn = (size_t)n * D + (size_t)n + 2 * D;
    zero_ws_kernel<<<(unsigned)((zn + 511) / 512), 512, 0, stream>>>(agg, zn);

    // K1: edge scatter-add (one wave32 per edge)
    int blocksE = (E + 7) / 8;                      // 8 waves / 256-thread block
    edge_scatter_kernel<<<blocksE, 256, 0, stream>>>(x, src, dst, agg, deg, E);

    // K2: WMMA GEMM + L2 norm + column stats (64 nodes per block)
    int blocksG = (n + 16 * NT - 1) / (16 * NT);
    sage_gemm_kernel<<<blocksG, 256, 0, stream>>>(x, agg, deg, Wl, bl, Wr,
                                                  h, colsum, colsumsq, n);

    // K3: stats finalize
    gn_stats_kernel<<<1, D, 0, stream>>>(colsum, colsumsq, alpha, mean, invstd, n);

    // K4: GraphNorm + GELU + residual
    size_t total = (size_t)n * D;
    gn_gelu_res_kernel<<<(unsigned)((total + 255) / 256), 256, 0, stream>>>(
        x, h, gw, gb, alpha, mean, invstd, total);
}